// GIN_9216999817920
// MI455X (gfx1250) — compile-verified
//
#include <hip/hip_runtime.h>

typedef float v2f __attribute__((ext_vector_type(2)));
typedef float v8f __attribute__((ext_vector_type(8)));

#define BN_EPS 1e-5f

// ---------------------------------------------------------------- utility ---
__global__ __launch_bounds__(256) void copy_f4(const float4* __restrict__ in,
                                               float4* __restrict__ out, long n) {
  long i = (long)blockIdx.x * blockDim.x + threadIdx.x;
  if (i < n) out[i] = in[i];
}

__global__ __launch_bounds__(256) void zero_f4(float4* __restrict__ out, long n) {
  long i = (long)blockIdx.x * blockDim.x + threadIdx.x;
  if (i < n) out[i] = make_float4(0.f, 0.f, 0.f, 0.f);
}

// agg[dst] += h[src], per edge, 4 channels per thread (float4 gather + f32 atomics)
__global__ __launch_bounds__(256) void scatter_add(const float* __restrict__ h,
                                                   const int* __restrict__ srcs,
                                                   const int* __restrict__ dsts,
                                                   float* __restrict__ agg,
                                                   long nedge, int C) {
  const int c4 = C >> 2;
  long i = (long)blockIdx.x * blockDim.x + threadIdx.x;
  if (i >= nedge * (long)c4) return;
  long e = i / c4;
  int c = (int)(i - e * (long)c4) << 2;
  int s = srcs[e], d = dsts[e];
  const float4 v = *(const float4*)(h + (long)s * C + c);
  float* p = agg + (long)d * C + c;
  atomicAdd(p + 0, v.x); atomicAdd(p + 1, v.y);
  atomicAdd(p + 2, v.z); atomicAdd(p + 3, v.w);
}

// pooled[batch[n]] += h[n]
__global__ __launch_bounds__(256) void pool_add(const float* __restrict__ h,
                                                const int* __restrict__ batch,
                                                float* __restrict__ pooled,
                                                long nnode, int C) {
  const int c4 = C >> 2;
  long i = (long)blockIdx.x * blockDim.x + threadIdx.x;
  if (i >= nnode * (long)c4) return;
  long n = i / c4;
  int c = (int)(i - n * (long)c4) << 2;
  int g = batch[n];
  const float4 v = *(const float4*)(h + n * (long)C + c);
  float* p = pooled + (long)g * C + c;
  atomicAdd(p + 0, v.x); atomicAdd(p + 1, v.y);
  atomicAdd(p + 2, v.z); atomicAdd(p + 3, v.w);
}

// -------------------------------------------------------- fused GIN layer ---
// out[16 rows x 256] = relu( relu(BN( Y@w1 + b1 )) @ w2 + b2 )
// block: 256 threads = 8 waves; wave w owns output columns [32w, 32w+32)
// as TWO 16x16 WMMA tiles sharing one A fragment.
// h1 tile lives only in LDS; fp32 WMMA 16x16x4 for both GEMMs.
__global__ __launch_bounds__(256) void gin_layer(
    const float* __restrict__ y,    // [nrows, ci]  (x + neighbor sum)
    const float* __restrict__ w1,   // [ci, 256]
    const float* __restrict__ b1,   // [256]
    const float* __restrict__ gam, const float* __restrict__ bet,
    const float* __restrict__ mu,  const float* __restrict__ var,
    const float* __restrict__ w2,   // [256, 256]
    const float* __restrict__ b2,   // [256]
    float* __restrict__ out,        // [nrows, 256]
    int nrows, int ci) {
  constexpr int CO = 256;
  constexpr int SH = 260;                       // padded LDS stride (bank-safe)
  __shared__ __align__(16) float Ylds[16 * 260];
  __shared__ __align__(16) float Hlds[16 * 260];

  const int tid   = threadIdx.x;
  const int lane  = tid & 31;
  const int wave  = tid >> 5;
  const int rowb  = blockIdx.x * 16;
  const int sy    = ci + 4;
  const int mhalf = lane >> 4;                  // 0|1 selects K pair / M+8
  const int mrow  = lane & 15;                  // A row (M) for this lane
  const int col0  = wave * 32 + (lane & 15);    // first  output column
  const int col1  = col0 + 16;                  // second output column

  // cooperative load of the 16 x ci input tile
  const int nvec = (16 * ci) >> 2;
  for (int idx = tid; idx < nvec; idx += 256) {
    int r  = idx / (ci >> 2);
    int c  = (idx - r * (ci >> 2)) << 2;
    float4 v4 = make_float4(0.f, 0.f, 0.f, 0.f);
    if (rowb + r < nrows) v4 = *(const float4*)(y + (long)(rowb + r) * ci + c);
    *(float4*)(Ylds + r * sy + c) = v4;
  }
  __syncthreads();

  // ---- GEMM1: Y(16 x ci) @ w1(ci x 256), this wave's 2 x 16 columns ----
  v8f acc0 = {}, acc1 = {};
  for (int k0 = 0; k0 < ci; k0 += 4) {
    const int ka = k0 + 2 * mhalf;
    v2f a, b0, b1v;
    a.x  = Ylds[mrow * sy + ka];
    a.y  = Ylds[mrow * sy + ka + 1];
    b0.x = w1[(long)ka * CO + col0];
    b0.y = w1[(long)(ka + 1) * CO + col0];
    b1v.x = w1[(long)ka * CO + col1];
    b1v.y = w1[(long)(ka + 1) * CO + col1];
    acc0 = __builtin_amdgcn_wmma_f32_16x16x4_f32(false, a, false, b0,
                                                 (short)0, acc0, false, false);
    acc1 = __builtin_amdgcn_wmma_f32_16x16x4_f32(false, a, false, b1v,
                                                 (short)0, acc1, false, false);
  }

  // epilogue: bias + BN + ReLU -> LDS (h1 never leaves the WGP)
  {
    const float sc0 = gam[col0] * rsqrtf(var[col0] + BN_EPS);
    const float sh0 = bet[col0] - mu[col0] * sc0;
    const float bb0 = b1[col0];
    const float sc1 = gam[col1] * rsqrtf(var[col1] + BN_EPS);
    const float sh1 = bet[col1] - mu[col1] * sc1;
    const float bb1 = b1[col1];
#pragma unroll
    for (int r = 0; r < 8; ++r) {
      const int row = r + 8 * mhalf;
      Hlds[row * SH + col0] = fmaxf((acc0[r] + bb0) * sc0 + sh0, 0.f);
      Hlds[row * SH + col1] = fmaxf((acc1[r] + bb1) * sc1 + sh1, 0.f);
    }
  }
  __syncthreads();

  // ---- GEMM2: h1(16 x 256) @ w2(256 x 256) ----
  v8f acc2 = {}, acc3 = {};
  for (int k0 = 0; k0 < CO; k0 += 4) {
    const int ka = k0 + 2 * mhalf;
    v2f a, b0, b1v;
    a.x  = Hlds[mrow * SH + ka];
    a.y  = Hlds[mrow * SH + ka + 1];
    b0.x = w2[(long)ka * CO + col0];
    b0.y = w2[(long)(ka + 1) * CO + col0];
    b1v.x = w2[(long)ka * CO + col1];
    b1v.y = w2[(long)(ka + 1) * CO + col1];
    acc2 = __builtin_amdgcn_wmma_f32_16x16x4_f32(false, a, false, b0,
                                                 (short)0, acc2, false, false);
    acc3 = __builtin_amdgcn_wmma_f32_16x16x4_f32(false, a, false, b1v,
                                                 (short)0, acc3, false, false);
  }
  const float bb2 = b2[col0];
  const float bb3 = b2[col1];
#pragma unroll
  for (int r = 0; r < 8; ++r) {
    int m = rowb + r + 8 * mhalf;
    if (m < nrows) {
      out[(long)m * CO + col0] = fmaxf(acc2[r] + bb2, 0.f);
      out[(long)m * CO + col1] = fmaxf(acc3[r] + bb3, 0.f);
    }
  }
}

// ------------------------------------------------------------- fused head ---
// out[16] = relu(BN(pooled(16x256) @ l0w + l0b)) @ l1w + l1b
// CO = 128 -> 8 waves cover all 8 column tiles exactly.
__global__ __launch_bounds__(256) void head_kernel(
    const float* __restrict__ pooled,  // [G, 256]
    const float* __restrict__ l0w,     // [256, 128]
    const float* __restrict__ l0b,
    const float* __restrict__ gam, const float* __restrict__ bet,
    const float* __restrict__ mu,  const float* __restrict__ var,
    const float* __restrict__ l1w,     // [128, 1]
    const float* __restrict__ l1b,     // [1]
    float* __restrict__ out, int ngraph) {
  constexpr int CI = 256, CO = 128;
  constexpr int SY = 260, SH = 132;
  __shared__ __align__(16) float Ylds[16 * 260];
  __shared__ __align__(16) float Hlds[16 * 132];

  const int tid   = threadIdx.x;
  const int lane  = tid & 31;
  const int wave  = tid >> 5;
  const int rowb  = blockIdx.x * 16;
  const int mhalf = lane >> 4;
  const int mrow  = lane & 15;
  const int col   = wave * 16 + (lane & 15);   // 8 waves * 16 = 128 cols

  for (int idx = tid; idx < (16 * CI) >> 2; idx += 256) {
    int r = idx >> 6;
    int c = (idx & 63) << 2;
    float4 v4 = make_float4(0.f, 0.f, 0.f, 0.f);
    if (rowb + r < ngraph) v4 = *(const float4*)(pooled + (long)(rowb + r) * CI + c);
    *(float4*)(Ylds + r * SY + c) = v4;
  }
  __syncthreads();

  v8f acc = {};
  for (int k0 = 0; k0 < CI; k0 += 4) {
    const int ka = k0 + 2 * mhalf;
    v2f a, b;
    a.x = Ylds[mrow * SY + ka];
    a.y = Ylds[mrow * SY + ka + 1];
    b.x = l0w[(long)ka * CO + col];
    b.y = l0w[(long)(ka + 1) * CO + col];
    acc = __builtin_amdgcn_wmma_f32_16x16x4_f32(false, a, false, b,
                                                (short)0, acc, false, false);
  }
  const float sc = gam[col] * rsqrtf(var[col] + BN_EPS);
  const float sh = bet[col] - mu[col] * sc;
  const float bias = l0b[col];
#pragma unroll
  for (int r = 0; r < 8; ++r) {
    float hv = (acc[r] + bias) * sc + sh;
    Hlds[(r + 8 * mhalf) * SH + col] = fmaxf(hv, 0.f);
  }
  __syncthreads();

  if (tid < 16 && rowb + tid < ngraph) {
    float a = l1b[0];
    for (int k = 0; k < CO; ++k) a += Hlds[tid * SH + k] * l1w[k];
    out[rowb + tid] = a;
  }
}

// ------------------------------------------------------------------ launch --
extern "C" void kernel_launch(void* const* d_in, const int* in_sizes, int n_in,
                              void* d_out, int out_size, void* d_ws, size_t ws_size,
                              hipStream_t stream) {
  const int C0 = 128, C = 256;
  const long N = in_sizes[0] / C0;
  const long E = in_sizes[1] / 2;
  const int  G = out_size;

  const float* x     = (const float*)d_in[0];
  const int*   ei    = (const int*)d_in[1];
  const int*   batch = (const int*)d_in[2];
  const int*   src   = ei;
  const int*   dst   = ei + E;

  // conv params start at index 3, 8 per layer
  auto P = [&](int layer, int j) -> const float* {
    return (const float*)d_in[3 + 8 * layer + j];
  };
  const float* l0w  = (const float*)d_in[27];
  const float* l0b  = (const float*)d_in[28];
  const float* l0g  = (const float*)d_in[29];
  const float* l0be = (const float*)d_in[30];
  const float* l0m  = (const float*)d_in[31];
  const float* l0v  = (const float*)d_in[32];
  const float* l1w  = (const float*)d_in[33];
  const float* l1b  = (const float*)d_in[34];

  float* buf0   = (float*)d_ws;              // y = h + agg(h)   [N*256]
  float* buf1   = buf0 + N * (long)C;        // layer output h   [N*256]
  float* pooled = buf1 + N * (long)C;        // [G*256]

  const int T = 256;
  auto blocks = [](long n) { return (unsigned)((n + 255) / 256); };
  const int rowTiles   = (int)((N + 15) / 16);
  const int graphTiles = (G + 15) / 16;

  // ---- layer 0 (ci = 128) ----
  copy_f4<<<blocks(N * C0 / 4), T, 0, stream>>>((const float4*)x, (float4*)buf0, N * C0 / 4);
  scatter_add<<<blocks(E * (C0 / 4)), T, 0, stream>>>(x, src, dst, buf0, E, C0);
  gin_layer<<<rowTiles, T, 0, stream>>>(buf0, P(0,0), P(0,1), P(0,2), P(0,3),
                                        P(0,4), P(0,5), P(0,6), P(0,7),
                                        buf1, (int)N, C0);
  // ---- layers 1,2 (ci = 256) ----
  for (int l = 1; l < 3; ++l) {
    copy_f4<<<blocks(N * C / 4), T, 0, stream>>>((const float4*)buf1, (float4*)buf0, N * C / 4);
    scatter_add<<<blocks(E * (C / 4)), T, 0, stream>>>(buf1, src, dst, buf0, E, C);
    gin_layer<<<rowTiles, T, 0, stream>>>(buf0, P(l,0), P(l,1), P(l,2), P(l,3),
                                          P(l,4), P(l,5), P(l,6), P(l,7),
                                          buf1, (int)N, C);
  }
  // ---- global_add_pool + head ----
  zero_f4<<<blocks((long)G * C / 4), T, 0, stream>>>((float4*)pooled, (long)G * C / 4);
  pool_add<<<blocks(N * (C / 4)), T, 0, stream>>>(buf1, batch, pooled, N, C);
  head_kernel<<<graphTiles, T, 0, stream>>>(pooled, l0w, l0b, l0g, l0be, l0m, l0v,
                                            l1w, l1b, (float*)d_out, G);
}